// Encoder_5188320493795
// MI455X (gfx1250) — compile-verified
//
#include <hip/hip_runtime.h>
#include <hip/hip_bf16.h>

typedef float v2f __attribute__((ext_vector_type(2)));
typedef float v8f __attribute__((ext_vector_type(8)));

#define SCAN_T 256

// ---------------- CSR build ----------------

__global__ void k_zero_i32(int* __restrict__ p, int n) {
  int i = blockIdx.x * blockDim.x + threadIdx.x;
  if (i < n) p[i] = 0;
}

__global__ void k_count(const int* __restrict__ dst, int* __restrict__ counts, int E) {
  int i = blockIdx.x * blockDim.x + threadIdx.x;
  if (i < E) atomicAdd(&counts[dst[i]], 1);
}

__global__ void k_scan_local(const int* __restrict__ counts, int* __restrict__ offsets,
                             int* __restrict__ blockSums, int n) {
  __shared__ int s[SCAN_T];
  const int tid = threadIdx.x;
  const int i = blockIdx.x * SCAN_T + tid;
  int v = (i < n) ? counts[i] : 0;
  s[tid] = v;
  __syncthreads();
  for (int d = 1; d < SCAN_T; d <<= 1) {
    int t = (tid >= d) ? s[tid - d] : 0;
    __syncthreads();
    s[tid] += t;
    __syncthreads();
  }
  if (i < n) offsets[i] = s[tid] - v;                 // block-local exclusive
  if (tid == SCAN_T - 1) blockSums[blockIdx.x] = s[tid];
}

__global__ void k_scan_blocks(int* __restrict__ blockSums, int nb) {
  __shared__ int s[SCAN_T];
  const int tid = threadIdx.x;
  int v = (tid < nb) ? blockSums[tid] : 0;
  s[tid] = v;
  __syncthreads();
  for (int d = 1; d < SCAN_T; d <<= 1) {
    int t = (tid >= d) ? s[tid - d] : 0;
    __syncthreads();
    s[tid] += t;
    __syncthreads();
  }
  if (tid < nb) blockSums[tid] = s[tid] - v;          // exclusive
}

__global__ void k_scan_add(int* __restrict__ offsets, const int* __restrict__ blockSums,
                           int* __restrict__ cursor, int n, int E) {
  int i = blockIdx.x * blockDim.x + threadIdx.x;
  if (i < n) {
    int o = offsets[i] + blockSums[i / SCAN_T];
    offsets[i] = o;
    cursor[i] = o;
  }
  if (i == 0) offsets[n] = E;
}

__global__ void k_bucket(const int* __restrict__ dst, int* __restrict__ cursor,
                         int* __restrict__ perm, int E) {
  int i = blockIdx.x * blockDim.x + threadIdx.x;
  if (i < E) {
    int slot = atomicAdd(&cursor[dst[i]], 1);
    perm[slot] = i;
  }
}

// Sort each destination bucket by edge id so FP accumulation order is
// deterministic across calls (buckets average E/N = 16 elements).
__global__ void k_sort_buckets(const int* __restrict__ offsets, int* __restrict__ perm, int n) {
  int d = blockIdx.x * blockDim.x + threadIdx.x;
  if (d >= n) return;
  const int beg = offsets[d], end = offsets[d + 1];
  for (int i = beg + 1; i < end; ++i) {
    int key = perm[i];
    int j = i - 1;
    while (j >= beg && perm[j] > key) { perm[j + 1] = perm[j]; --j; }
    perm[j + 1] = key;
  }
}

// ---------------- GEMM: C[M,N] = A[M,K] @ B[K,N] via V_WMMA_F32_16X16X4_F32 ----
// grid.x = M/16 tiles; blockDim = (N/16)*32 threads; wave w owns N-tile w.
// A layout (ISA 7.12.2, 32-bit A 16x4): lane&15 = m, lane[4] selects K-pair
//   {0,1} vs {2,3}; v0 = first K of pair, v1 = second.
// B layout (4x16, rows striped across lanes): lane&15 = n, same K-pair select.
// C layout (32-bit 16x16): vgpr r -> m = r + 8*lane[4], n = lane&15.

__global__ void __launch_bounds__(256)
k_gemm_wmma_f32(const float* __restrict__ A, const float* __restrict__ B,
                float* __restrict__ C, int M, int K, int N) {
  const int lane = threadIdx.x & 31;
  const int wave = threadIdx.x >> 5;
  const int tm = blockIdx.x * 16;
  const int tn = wave * 16;
  if (tn >= N) return;
  const int ln = lane & 15;            // m for A, n for B/C
  const int kb = (lane >> 4) << 1;     // 0 or 2
  const float* arow = A + (size_t)min(tm + ln, M - 1) * K;
  const float* bcol = B + tn + ln;
  v8f acc = {};
  for (int k0 = 0; k0 < K; k0 += 4) {
    v2f a, b;
    a.x = arow[k0 + kb];
    a.y = arow[k0 + kb + 1];
    b.x = bcol[(size_t)(k0 + kb) * N];
    b.y = bcol[(size_t)(k0 + kb + 1) * N];
    acc = __builtin_amdgcn_wmma_f32_16x16x4_f32(
        /*neg_a=*/false, a, /*neg_b=*/false, b,
        /*c_mod=*/(short)0, acc, /*reuse_a=*/false, /*reuse_b=*/false);
  }
  const int mbase = (lane >> 4) << 3;  // 0 or 8
#pragma unroll
  for (int r = 0; r < 8; ++r) {
    int m = tm + mbase + r;
    if (m < M) C[(size_t)m * N + tn + ln] = acc[r];
  }
}

// ---------------- Per-destination gather ----------------
// out[d] = relu(sum_e w_e * support[src_e] + b), one wave32 per destination row,
// each lane holds 4 of the 128 features (float4).
//
// Edge metadata is loaded 32-edges-at-a-time (one coalesced load round: lane j
// fetches edge base+j's perm/weight/src), then broadcast with __shfl so the
// inner loop's only memory ops are the independent 512B support-row gathers —
// no dependent-load chain, and the L2-resident gathers can pipeline deeply.

__global__ void __launch_bounds__(256)
k_gather(const float* __restrict__ support, const int* __restrict__ offsets,
         const int* __restrict__ perm, const int* __restrict__ esrc,
         const float* __restrict__ ew, const float* __restrict__ bias,
         float* __restrict__ out, int n) {
  const int lane = threadIdx.x & 31;
  const int wave = threadIdx.x >> 5;
  const int d = blockIdx.x * (blockDim.x >> 5) + wave;
  if (d >= n) return;
  const int beg = offsets[d], end = offsets[d + 1];
  float4 acc = make_float4(0.f, 0.f, 0.f, 0.f);
  const float4* sup4 = (const float4*)support;

  for (int base = beg; base < end; base += 32) {
    const int idx = base + lane;
    const int e = perm[(idx < end) ? idx : beg];   // clamp: safe, unused lanes ignored
    const float w = ew[e];
    const int s = esrc[e];
    const int cnt = min(end - base, 32);
#pragma unroll 4
    for (int j = 0; j < cnt; ++j) {
      const float wj = __shfl(w, j, 32);
      const int sj = __shfl(s, j, 32);
      const float4 v = sup4[(size_t)sj * 32 + lane];
      acc.x = fmaf(wj, v.x, acc.x);
      acc.y = fmaf(wj, v.y, acc.y);
      acc.z = fmaf(wj, v.z, acc.z);
      acc.w = fmaf(wj, v.w, acc.w);
    }
  }

  const float4 b = ((const float4*)bias)[lane];
  acc.x = fmaxf(acc.x + b.x, 0.f);
  acc.y = fmaxf(acc.y + b.y, 0.f);
  acc.z = fmaxf(acc.z + b.z, 0.f);
  acc.w = fmaxf(acc.w + b.w, 0.f);
  ((float4*)out)[(size_t)d * 32 + lane] = acc;
}

// ---------------- driver ----------------

extern "C" void kernel_launch(void* const* d_in, const int* in_sizes, int n_in,
                              void* d_out, int out_size, void* d_ws, size_t ws_size,
                              hipStream_t stream) {
  const float* x   = (const float*)d_in[0];
  const int* esrc  = (const int*)d_in[1];
  const int* edst  = (const int*)d_in[2];
  const float* ew  = (const float*)d_in[3];
  const float* W1  = (const float*)d_in[4];
  const float* b1  = (const float*)d_in[5];
  const float* W2  = (const float*)d_in[6];
  const float* b2  = (const float*)d_in[7];

  const int E     = in_sizes[1];
  const int nhid  = in_sizes[5];           // 128
  const int nfeat = in_sizes[4] / nhid;    // 256
  const int N     = in_sizes[0] / nfeat;   // 50000

  // workspace layout
  char* ws = (char*)d_ws;
  size_t off = 0;
  auto alloc = [&](size_t bytes) -> void* {
    void* p = ws + off;
    off += bytes;
    off = (off + 255) & ~(size_t)255;
    return p;
  };
  float* support = (float*)alloc((size_t)N * nhid * sizeof(float)); // reused for both layers
  float* h       = (float*)alloc((size_t)N * nhid * sizeof(float));
  int* offsets   = (int*)alloc((size_t)(N + 1) * sizeof(int));
  int* cnt_cur   = (int*)alloc((size_t)N * sizeof(int));            // counts, then cursor
  int* perm      = (int*)alloc((size_t)E * sizeof(int));
  int* blockSums = (int*)alloc(1024 * sizeof(int));
  (void)ws_size; (void)n_in; (void)out_size;

  const int T = 256;
  const int nbScan = (N + SCAN_T - 1) / SCAN_T;   // 196 <= 256

  // Build destination-CSR (shared by both layers)
  k_zero_i32<<<(N + T - 1) / T, T, 0, stream>>>(cnt_cur, N);
  k_count<<<(E + T - 1) / T, T, 0, stream>>>(edst, cnt_cur, E);
  k_scan_local<<<nbScan, SCAN_T, 0, stream>>>(cnt_cur, offsets, blockSums, N);
  k_scan_blocks<<<1, SCAN_T, 0, stream>>>(blockSums, nbScan);
  k_scan_add<<<(N + T - 1) / T, T, 0, stream>>>(offsets, blockSums, cnt_cur, N, E);
  k_bucket<<<(E + T - 1) / T, T, 0, stream>>>(edst, cnt_cur, perm, E);
  k_sort_buckets<<<(N + T - 1) / T, T, 0, stream>>>(offsets, perm, N);

  const int gemmBlock = (nhid / 16) * 32;         // 256 threads = 8 waves
  const int gemmGrid  = (N + 15) / 16;            // 3125 M-tiles

  // Layer 1: support = x @ W1 ; h = relu(A @ support + b1)
  k_gemm_wmma_f32<<<gemmGrid, gemmBlock, 0, stream>>>(x, W1, support, N, nfeat, nhid);
  k_gather<<<(N + 7) / 8, 256, 0, stream>>>(support, offsets, perm, esrc, ew, b1, h, N);

  // Layer 2: support = h @ W2 ; out = relu(A @ support + b2)
  k_gemm_wmma_f32<<<gemmGrid, gemmBlock, 0, stream>>>(h, W2, support, N, nhid, nhid);
  k_gather<<<(N + 7) / 8, 256, 0, stream>>>(support, offsets, perm, esrc, ew, b2,
                                            (float*)d_out, N);
}